// GraphConv_GRU_30734785970522
// MI455X (gfx1250) — compile-verified
//
#include <hip/hip_runtime.h>
#include <math.h>

#define N_NODES 100000
#define N_EDGES 1600000
#define F 128
#define NT (N_NODES / 16)   // 6250 tiles of 16 nodes
#define WPB 8               // waves (tiles) per block in the GEMM kernel

typedef __attribute__((ext_vector_type(2))) float v2f;
typedef __attribute__((ext_vector_type(8))) float v8f;

// ---------------- small utility kernels ----------------

__global__ void zero_f32(float* __restrict__ p, int n) {
    int i = blockIdx.x * blockDim.x + threadIdx.x;
    if (i < n) p[i] = 0.0f;
}

__global__ void degree_kernel(const int* __restrict__ dst, float* __restrict__ deg) {
    int e = blockIdx.x * blockDim.x + threadIdx.x;
    if (e < N_EDGES) atomicAdd(&deg[dst[e]], 1.0f);
}

__global__ void norm_kernel(const float* __restrict__ deg, float* __restrict__ norm) {
    int i = blockIdx.x * blockDim.x + threadIdx.x;
    if (i < N_NODES) norm[i] = rsqrtf(fmaxf(deg[i], 1.0f));
}

// wT[n][k] = weight[k][n]  (so phase-1 B fragments are contiguous along K)
__global__ void transpose_weight(const float* __restrict__ w, float* __restrict__ wT) {
    int t = blockIdx.x * blockDim.x + threadIdx.x;
    if (t < F * F) {
        int k = t >> 7;        // t / F
        int n = t & (F - 1);   // t % F
        wT[(size_t)n * F + k] = w[t];
    }
}

// featn[i,:] = feat[i,:] * norm[i]   (float4 per thread: N*32 threads)
__global__ void featn_kernel(const float* __restrict__ feat,
                             const float* __restrict__ norm,
                             float* __restrict__ featn) {
    int t = blockIdx.x * blockDim.x + threadIdx.x;
    if (t >= N_NODES * 32) return;
    int row = t >> 5;
    int c4  = (t & 31) << 2;
    float nv = norm[row];
    float4 v = *(const float4*)(feat + (size_t)row * F + c4);
    v.x *= nv; v.y *= nv; v.z *= nv; v.w *= nv;
    *(float4*)(featn + (size_t)row * F + c4) = v;
}

// agg[dst[e],:] += featn[src[e],:]   one wave per edge, float4 per lane
__global__ void scatter_kernel(const int* __restrict__ src, const int* __restrict__ dst,
                               const float* __restrict__ featn, float* __restrict__ agg) {
    int t = blockIdx.x * blockDim.x + threadIdx.x;   // E*32 threads
    int e = t >> 5;
    if (e >= N_EDGES) return;
    int lane = t & 31;
    int s = src[e], d = dst[e];
    float4 v = *(const float4*)(featn + (size_t)s * F + lane * 4);
    float* a = agg + (size_t)d * F + lane * 4;
    atomicAdd(a + 0, v.x);
    atomicAdd(a + 1, v.y);
    atomicAdd(a + 2, v.z);
    atomicAdd(a + 3, v.w);
}

// ---------------- fused GEMM (f32 WMMA) + GRU kernel ----------------
//
// One wave handles a 16-node tile.
// Phase 1: rst = (agg @ weight) * norm + bias  -> per-wave LDS slab (16x128 f32)
//          A tile held in 64 VGPRs (v2f aA[32]), B from pre-transposed wT.
// Phase 2: gi = featn @ w_ih^T, gh = rst @ w_hh^T, then GRU gating -> out
//
// f32 WMMA fragment layout (16x16x4):
//   A: lane(m = lane&15, half = lane>>4) holds {A[m][2h], A[m][2h+1]} in vgpr {0,1}
//   B: lane(n = lane&15, half)           holds {B[2h][n], B[2h+1][n]}
//   C/D: vgpr r <-> row m = r + 8*half, col n = lane&15

__global__ __launch_bounds__(256)
void gemm_gru_kernel(const float* __restrict__ agg, const float* __restrict__ featn,
                     const float* __restrict__ norm,
                     const float* __restrict__ wT, const float* __restrict__ bias,
                     const float* __restrict__ w_ih, const float* __restrict__ w_hh,
                     const float* __restrict__ b_ih, const float* __restrict__ b_hh,
                     float* __restrict__ out) {
    __shared__ float rst_lds[WPB * 16 * F];   // 64 KB: one 16x128 f32 slab per wave

    const int wave = threadIdx.x >> 5;
    const int lane = threadIdx.x & 31;
    const int tile = blockIdx.x * WPB + wave;
    if (tile >= NT) return;                    // wave-uniform: EXEC stays all-ones

    const int rowBase = tile * 16;
    const int half    = lane >> 4;
    const int mr      = lane & 15;             // A row / B col / D col for this lane
    float* rstL = &rst_lds[wave * 16 * F];

    // per-lane row norms for the D-layout epilogues (rows m = r + 8*half)
    float nrm[8];
#pragma unroll
    for (int r = 0; r < 8; ++r) nrm[r] = norm[rowBase + r + 8 * half];

    // ---------- Phase 1: rst = (agg @ weight) * norm + bias ----------
    // Hoist the whole 16x128 A tile into registers (read agg exactly once).
    v2f aA[32];
#pragma unroll
    for (int kk = 0; kk < 32; ++kk)
        aA[kk] = *(const v2f*)(agg + (size_t)(rowBase + mr) * F + kk * 4 + 2 * half);

    for (int c = 0; c < 8; ++c) {
        const int colBase = c * 16;
        const float* bcol = wT + (size_t)(colBase + mr) * F + 2 * half;
        v8f acc = {};
#pragma unroll
        for (int kk = 0; kk < 32; ++kk) {
            v2f b = *(const v2f*)(bcol + kk * 4);
            acc = __builtin_amdgcn_wmma_f32_16x16x4_f32(
                      false, aA[kk], false, b, (short)0, acc, false, false);
        }
        const float bv = bias[colBase + mr];
#pragma unroll
        for (int r = 0; r < 8; ++r) {
            const int m = r + 8 * half;
            rstL[m * F + colBase + mr] = acc[r] * nrm[r] + bv;
        }
    }
    // (per-wave LDS slab; same wave reads it back -> in-order DS, no barrier needed)

    // ---------- Phase 2: GRU gates ----------
    for (int c = 0; c < 8; ++c) {
        const int colBase = c * 16;
        const int nrow = colBase + mr;           // row of w_ih / w_hh (transposed use)
        const float* pbr = w_ih + (size_t)(nrow)         * F + 2 * half;
        const float* pbz = w_ih + (size_t)(nrow + F)     * F + 2 * half;
        const float* pbn = w_ih + (size_t)(nrow + 2 * F) * F + 2 * half;
        const float* pcr = w_hh + (size_t)(nrow)         * F + 2 * half;
        const float* pcz = w_hh + (size_t)(nrow + F)     * F + 2 * half;
        const float* pcn = w_hh + (size_t)(nrow + 2 * F) * F + 2 * half;
        const float* paf = featn + (size_t)(rowBase + mr) * F + 2 * half;
        const float* pah = rstL + mr * F + 2 * half;

        v8f ir = {}, iz = {}, in_ = {}, hr = {}, hz = {}, hn = {};
        for (int k0 = 0; k0 < F; k0 += 4) {
            v2f af = *(const v2f*)(paf + k0);
            v2f ah = *(const v2f*)(pah + k0);
            v2f br = *(const v2f*)(pbr + k0);
            v2f bz = *(const v2f*)(pbz + k0);
            v2f bn = *(const v2f*)(pbn + k0);
            v2f cr = *(const v2f*)(pcr + k0);
            v2f cz = *(const v2f*)(pcz + k0);
            v2f cn = *(const v2f*)(pcn + k0);
            ir  = __builtin_amdgcn_wmma_f32_16x16x4_f32(false, af, false, br, (short)0, ir,  false, false);
            iz  = __builtin_amdgcn_wmma_f32_16x16x4_f32(false, af, false, bz, (short)0, iz,  false, false);
            in_ = __builtin_amdgcn_wmma_f32_16x16x4_f32(false, af, false, bn, (short)0, in_, false, false);
            hr  = __builtin_amdgcn_wmma_f32_16x16x4_f32(false, ah, false, cr, (short)0, hr,  false, false);
            hz  = __builtin_amdgcn_wmma_f32_16x16x4_f32(false, ah, false, cz, (short)0, hz,  false, false);
            hn  = __builtin_amdgcn_wmma_f32_16x16x4_f32(false, ah, false, cn, (short)0, hn,  false, false);
        }
#pragma unroll
        for (int r = 0; r < 8; ++r) {
            const int m   = r + 8 * half;
            const int col = colBase + mr;
            float gir = ir[r]  + b_ih[col];
            float giz = iz[r]  + b_ih[F + col];
            float gin = in_[r] + b_ih[2 * F + col];
            float ghr = hr[r]  + b_hh[col];
            float ghz = hz[r]  + b_hh[F + col];
            float ghn = hn[r]  + b_hh[2 * F + col];
            float rg = 1.0f / (1.0f + expf(-(gir + ghr)));
            float zg = 1.0f / (1.0f + expf(-(giz + ghz)));
            float ng = tanhf(gin + rg * ghn);
            float rstv = rstL[m * F + col];
            out[(size_t)(rowBase + m) * F + col] = (1.0f - zg) * ng + zg * rstv;
        }
    }
}

// ---------------- launcher ----------------

extern "C" void kernel_launch(void* const* d_in, const int* in_sizes, int n_in,
                              void* d_out, int out_size, void* d_ws, size_t ws_size,
                              hipStream_t stream) {
    const float* feat   = (const float*)d_in[0];
    const float* weight = (const float*)d_in[1];
    const float* bias   = (const float*)d_in[2];
    const float* w_ih   = (const float*)d_in[3];
    const float* w_hh   = (const float*)d_in[4];
    const float* b_ih   = (const float*)d_in[5];
    const float* b_hh   = (const float*)d_in[6];
    const int*   src    = (const int*)d_in[7];
    const int*   dst    = (const int*)d_in[8];
    float* out = (float*)d_out;

    // workspace layout (floats): deg[N] | norm[N] | featn[N*F] | agg[N*F] | wT[F*F]
    float* ws    = (float*)d_ws;
    float* deg   = ws;
    float* norm  = ws + N_NODES;
    float* featn = ws + 2 * N_NODES;
    float* agg   = featn + (size_t)N_NODES * F;
    float* wT    = agg + (size_t)N_NODES * F;

    const int B = 256;

    // zero deg and agg (workspace is poisoned; must re-zero every call)
    zero_f32<<<(N_NODES + B - 1) / B, B, 0, stream>>>(deg, N_NODES);
    zero_f32<<<(N_NODES * F + B - 1) / B, B, 0, stream>>>(agg, N_NODES * F);

    // degree -> norm -> featn; transpose weight for contiguous B fragments
    degree_kernel<<<(N_EDGES + B - 1) / B, B, 0, stream>>>(dst, deg);
    norm_kernel<<<(N_NODES + B - 1) / B, B, 0, stream>>>(deg, norm);
    transpose_weight<<<(F * F + B - 1) / B, B, 0, stream>>>(weight, wT);
    featn_kernel<<<(N_NODES * 32 + B - 1) / B, B, 0, stream>>>(feat, norm, featn);

    // SpMM scatter-add: one wave per edge
    scatter_kernel<<<(N_EDGES * 32 + B - 1) / B, B, 0, stream>>>(src, dst, featn, agg);

    // fused WMMA GEMMs + GRU gating
    gemm_gru_kernel<<<(NT + WPB - 1) / WPB, B, 0, stream>>>(
        agg, featn, norm, wT, bias, w_ih, w_hh, b_ih, b_hh, out);
}